// ToPMoE_68049461838327
// MI455X (gfx1250) — compile-verified
//
#include <hip/hip_runtime.h>
#include <hip/hip_bf16.h>

typedef __attribute__((ext_vector_type(2))) float v2f;
typedef __attribute__((ext_vector_type(8))) float v8f;
typedef int v4i __attribute__((vector_size(4 * sizeof(int))));

#define GLOBAL_AS __attribute__((address_space(1)))
#define LDS_AS    __attribute__((address_space(3)))

#if __has_builtin(__builtin_amdgcn_global_load_async_to_lds_b128)
#define HAVE_ASYNC 1
#else
#define HAVE_ASYNC 0
#endif

#if HAVE_ASYNC
#define ASYNC_COPY16(gp, lp)                                         \
  __builtin_amdgcn_global_load_async_to_lds_b128(                    \
      (GLOBAL_AS v4i*)(gp), (LDS_AS v4i*)(lp), 0, 0)
#if __has_builtin(__builtin_amdgcn_s_wait_asynccnt)
#define WAIT_ASYNC(n) __builtin_amdgcn_s_wait_asynccnt(n)
#else
#define WAIT_ASYNC(n) asm volatile("s_wait_asynccnt " #n ::: "memory")
#endif
#else
#define WAIT_ASYNC(n)
#endif

// ---------------------------------------------------------------------------
// Kernel 1: gating — logits = x @ Wg + bg, softmax over E=16, top-2.
// ---------------------------------------------------------------------------
__global__ __launch_bounds__(256) void moe_gate_topk(
    const float* __restrict__ x, const float* __restrict__ Wg,
    const float* __restrict__ bg, float* __restrict__ topv,
    int* __restrict__ topi, int* __restrict__ counts, int B, int D) {
  constexpr int E = 16;
  __shared__ float part[E * E];
  __shared__ float logits[E];
  const int b = blockIdx.x;
  const int t = threadIdx.x;
  const int e = t & 15;
  const int c = t >> 4;
  const float* xb = x + (size_t)b * D;
  float acc = 0.f;
  for (int d = c; d < D; d += 16) acc += xb[d] * Wg[d * E + e];
  part[c * E + e] = acc;
  __syncthreads();
  if (t < E) {
    float s = bg[t];
#pragma unroll
    for (int c2 = 0; c2 < E; ++c2) s += part[c2 * E + t];
    logits[t] = s;
  }
  __syncthreads();
  if (t == 0) {
    int i0 = 0; float v0 = logits[0];
#pragma unroll
    for (int i = 1; i < E; ++i) { if (logits[i] > v0) { v0 = logits[i]; i0 = i; } }
    int i1 = (i0 == 0) ? 1 : 0; float v1 = logits[i1];
#pragma unroll
    for (int i = 0; i < E; ++i) { if (i != i0 && logits[i] > v1) { v1 = logits[i]; i1 = i; } }
    float denom = 0.f;
#pragma unroll
    for (int i = 0; i < E; ++i) denom += __expf(logits[i] - v0);
    const float inv = 1.f / denom;
    topv[b * 2 + 0] = inv;                      // exp(v0 - v0) = 1
    topv[b * 2 + 1] = __expf(v1 - v0) * inv;
    topi[b * 2 + 0] = i0;
    topi[b * 2 + 1] = i1;
    atomicAdd(&counts[i0], 1);
    atomicAdd(&counts[i1], 1);
  }
}

__global__ void moe_init(float* __restrict__ zpad, int* __restrict__ counts) {
  const int t = threadIdx.x;
  if (t < 16) counts[t] = 0;
  else if (t < 32) zpad[t - 16] = 0.f;
}

__global__ void moe_scan(const int* __restrict__ counts,
                         int* __restrict__ offsets, int* __restrict__ cursors) {
  if (threadIdx.x == 0) {
    int run = 0;
#pragma unroll
    for (int e = 0; e < 16; ++e) {
      offsets[e] = run;
      run += counts[e];
      cursors[e] = 0;
    }
  }
}

__global__ __launch_bounds__(256) void moe_scatter(
    const int* __restrict__ topi, const int* __restrict__ offsets,
    int* __restrict__ cursors, int* __restrict__ list, int B) {
  const int b = blockIdx.x * 256 + threadIdx.x;
  if (b >= B) return;
#pragma unroll
  for (int k = 0; k < 2; ++k) {
    const int e = topi[b * 2 + k];
    const int pos = atomicAdd(&cursors[e], 1);
    list[offsets[e] + pos] = (b << 1) | k;
  }
}

__global__ __launch_bounds__(256) void moe_zero(float4* __restrict__ out, int n4) {
  const int i = blockIdx.x * 256 + threadIdx.x;
  if (i < n4) out[i] = make_float4(0.f, 0.f, 0.f, 0.f);
}

// ---------------------------------------------------------------------------
// Kernel 5: grouped expert GEMM, V_WMMA_F32_16X16X4_F32.
// Block = 256 threads (8 wave32), block tile 128 tokens x 256 cols.
// Wave grid 2(M) x 4(N); wave tile 4x4 WMMA tiles (64x64) -> 128 acc VGPRs.
// K staged in double-buffered LDS chunks of 32 via async global->LDS copies.
// ---------------------------------------------------------------------------
__global__ __launch_bounds__(256) void moe_expert_gemm(
    const float* __restrict__ x, const float* __restrict__ We,
    const float* __restrict__ be, const float* __restrict__ topv,
    const int* __restrict__ list, const int* __restrict__ counts,
    const int* __restrict__ offsets, const float* __restrict__ zpad,
    float* __restrict__ out, int B, int D) {
  constexpr int KC = 32;    // k-chunk
  constexpr int MT = 128;   // tokens per block
  constexpr int NC = 256;   // cols per block
  constexpr int ASZ = MT * KC;   // 4096 floats / buffer
  constexpr int BSZ = KC * NC;   // 8192 floats / buffer

  extern __shared__ __align__(16) char smem_raw[];
  float* AsBuf = (float*)smem_raw;        // [2][ASZ] : 32 KB
  float* BsBuf = AsBuf + 2 * ASZ;         // [2][BSZ] : 64 KB
  __shared__ int   toks[MT];
  __shared__ float wts[MT];

  const int e    = blockIdx.z;
  const int n_e  = counts[e];
  const int base = blockIdx.y * MT;
  if (base >= n_e) return;                // uniform per block
  const int col0 = blockIdx.x * NC;
  const int tid  = threadIdx.x;
  const int lane = tid & 31;
  const int wave = tid >> 5;
  const int l16  = lane & 15;
  const int kHalf   = (lane >> 4) << 1;   // 0 or 2 : K pair of this lane half
  const int rowHalf = (lane >> 4) << 3;   // 0 or 8 : M offset of C/D rows
  const int mBase = (wave >> 2) * 64;     // wave M group (2)
  const int nBase = (wave & 3) * 64;      // wave N group (4)

  if (tid < MT) {
    const int i = base + tid;
    if (i < n_e) {
      const int entry = list[offsets[e] + i];
      toks[tid] = entry >> 1;
      wts[tid]  = topv[entry];
    } else {
      toks[tid] = -1;
      wts[tid]  = 0.f;
    }
  }
  __syncthreads();

  const float* WeP = We + (size_t)e * D * D + col0;

  // ---- stage one (MT x KC) A chunk + (KC x NC) B chunk into buffer `buf` ----
  auto stage = [&](int buf, int k0) {
    float* As = AsBuf + buf * ASZ;
    float* Bs = BsBuf + buf * BSZ;
#pragma unroll
    for (int i = 0; i < ASZ / (4 * 256); ++i) {        // 4 x 16B per thread
      const int idx = tid + i * 256;
      const int row = idx >> 3;
      const int q   = (idx & 7) << 2;
      const int tk  = toks[row];
      const float* g = (tk >= 0) ? (x + (size_t)tk * D + k0 + q) : zpad;
#if HAVE_ASYNC
      ASYNC_COPY16(g, As + row * KC + q);
#else
      *(float4*)(As + row * KC + q) = *(const float4*)g;
#endif
    }
#pragma unroll
    for (int i = 0; i < BSZ / (4 * 256); ++i) {        // 8 x 16B per thread
      const int idx = tid + i * 256;
      const int row = idx >> 6;
      const int c4  = (idx & 63) << 2;
      const float* g = WeP + (size_t)(k0 + row) * D + c4;
#if HAVE_ASYNC
      ASYNC_COPY16(g, Bs + row * NC + c4);
#else
      *(float4*)(Bs + row * NC + c4) = *(const float4*)g;
#endif
    }
  };

  v8f acc[4][4];
#pragma unroll
  for (int m = 0; m < 4; ++m)
#pragma unroll
    for (int n = 0; n < 4; ++n)
      acc[m][n] = (v8f){0.f, 0.f, 0.f, 0.f, 0.f, 0.f, 0.f, 0.f};

  const int nch = D / KC;
  stage(0, 0);
  for (int kc = 0; kc < nch; ++kc) {
    const int cur = kc & 1;
    const bool hasNext = (kc + 1) < nch;
    if (hasNext) {
      stage(cur ^ 1, (kc + 1) * KC);
      WAIT_ASYNC(12);    // 12 per-wave async ops outstanding for next chunk
    } else {
      WAIT_ASYNC(0);
    }
    __syncthreads();     // all waves retired their chunk-kc copies -> buffer valid

    const float* As = AsBuf + cur * ASZ;
    const float* Bs = BsBuf + cur * BSZ;
#pragma unroll
    for (int kk = 0; kk < KC; kk += 4) {
      v2f aF[4];
      v2f bF[4];
#pragma unroll
      for (int m = 0; m < 4; ++m) {
        // A 16x4 layout: lane holds row l16; K = kHalf, kHalf+1
        const float* ap = &As[(mBase + m * 16 + l16) * KC + kk + kHalf];
        aF[m].x = ap[0];
        aF[m].y = ap[1];
      }
#pragma unroll
      for (int n = 0; n < 4; ++n) {
        // B 4x16 layout: lane holds col l16; K = kHalf, kHalf+1
        const int col = nBase + n * 16 + l16;
        bF[n].x = Bs[(kk + kHalf) * NC + col];
        bF[n].y = Bs[(kk + kHalf + 1) * NC + col];
      }
#pragma unroll
      for (int m = 0; m < 4; ++m)
#pragma unroll
        for (int n = 0; n < 4; ++n)
          acc[m][n] = __builtin_amdgcn_wmma_f32_16x16x4_f32(
              false, aF[m], false, bF[n], (short)0, acc[m][n], false, false);
    }
    __syncthreads();     // safe to overwrite this buffer with chunk kc+2
  }

  // ---- epilogue: out[token] += w * (acc + be[e]) via f32 atomics ----
#pragma unroll
  for (int m = 0; m < 4; ++m) {
#pragma unroll
    for (int r = 0; r < 8; ++r) {
      const int lrow = mBase + m * 16 + r + rowHalf;
      const int tk = toks[lrow];
      if (tk >= 0) {
        const float w = wts[lrow];
#pragma unroll
        for (int n = 0; n < 4; ++n) {
          const int col = col0 + nBase + n * 16 + l16;
          const float v = w * (acc[m][n][r] + be[e * D + col]);
          atomicAdd(&out[(size_t)tk * D + col], v);
        }
      }
    }
  }
}

// ---------------------------------------------------------------------------
extern "C" void kernel_launch(void* const* d_in, const int* in_sizes, int n_in,
                              void* d_out, int out_size, void* d_ws, size_t ws_size,
                              hipStream_t stream) {
  const float* x  = (const float*)d_in[0];
  const float* Wg = (const float*)d_in[1];
  const float* bg = (const float*)d_in[2];
  const float* We = (const float*)d_in[3];
  const float* be = (const float*)d_in[4];
  float* out = (float*)d_out;

  const int E = in_sizes[2];             // 16
  const int D = in_sizes[4] / E;         // 2048
  const int B = in_sizes[0] / D;         // 2048

  // workspace layout (all small)
  float* zpad    = (float*)d_ws;               // 16 floats, zeroed (async pad src)
  int*   counts  = (int*)(zpad + 16);
  int*   offsets = counts + 16;
  int*   cursors = offsets + 16;
  int*   list    = cursors + 16;               // 2*B entries
  float* topv    = (float*)(list + 2 * B);     // 2*B floats
  int*   topi    = (int*)(topv + 2 * B);       // 2*B ints

  moe_init<<<1, 64, 0, stream>>>(zpad, counts);
  moe_gate_topk<<<B, 256, 0, stream>>>(x, Wg, bg, topv, topi, counts, B, D);
  moe_scan<<<1, 32, 0, stream>>>(counts, offsets, cursors);
  moe_scatter<<<(B + 255) / 256, 256, 0, stream>>>(topi, offsets, cursors, list, B);

  const int n4 = (B * D) / 4;
  moe_zero<<<(n4 + 255) / 256, 256, 0, stream>>>((float4*)out, n4);

  constexpr int KC = 32, MT = 128, NC = 256;
  const size_t shmem = 2 * (size_t)(MT * KC + KC * NC) * sizeof(float);  // 96 KB
  dim3 grid(D / NC, (B + MT - 1) / MT, 16);    // (col slices, token chunks, experts)
  moe_expert_gemm<<<grid, 256, shmem, stream>>>(x, We, be, topv, list, counts,
                                                offsets, zpad, out, B, D);
}